// GMedPBlock_6193342841104
// MI455X (gfx1250) — compile-verified
//
#include <hip/hip_runtime.h>
#include <hip/hip_bf16.h>

typedef __attribute__((ext_vector_type(2))) float v2f;
typedef __attribute__((ext_vector_type(4))) float v4f;
typedef __attribute__((ext_vector_type(8))) float v8f;

// ---------------------------------------------------------------------------
// Kernel 1: per-row k-th order statistic via 4-pass in-register radix select.
// One 256-thread block per row (784 floats). x is read exactly once (NT loads).
// ---------------------------------------------------------------------------
__device__ __forceinline__ unsigned int f2ord(float f) {
  unsigned int u = __float_as_uint(f);
  // monotone map: float total order -> unsigned total order
  return u ^ ((u & 0x80000000u) ? 0xFFFFFFFFu : 0x80000000u);
}

__global__ __launch_bounds__(256) void kth_select_kernel(
    const float* __restrict__ x,   // [rows, N]
    float* __restrict__ med,       // [rows]
    int N, int kIdx) {
  const int row = blockIdx.x;
  const int t   = threadIdx.x;          // 0..255
  const float* xr = x + (size_t)row * N;

  // Each thread holds 4 keys in registers; pad with 0xFFFFFFFF (ranks last).
  unsigned int key[4];
  const int base = t * 4;
  if (base + 3 < N) {
    v4f v = __builtin_nontemporal_load((const v4f*)(xr + base));
    key[0] = f2ord(v.x);
    key[1] = f2ord(v.y);
    key[2] = f2ord(v.z);
    key[3] = f2ord(v.w);
  } else {
#pragma unroll
    for (int i = 0; i < 4; ++i)
      key[i] = (base + i < N) ? f2ord(xr[base + i]) : 0xFFFFFFFFu;
  }

  __shared__ unsigned int hist[256];
  __shared__ unsigned int scan[256];
  __shared__ unsigned int s_prefix;
  __shared__ unsigned int s_k;

  unsigned int prefix = 0u;
  unsigned int kRem   = (unsigned int)kIdx;

#pragma unroll
  for (int shift = 24; shift >= 0; shift -= 8) {
    hist[t] = 0u;
    __syncthreads();

    const unsigned int pmask =
        (shift == 24) ? 0u : (0xFFFFFFFFu << (shift + 8));
#pragma unroll
    for (int i = 0; i < 4; ++i) {
      const unsigned int kk = key[i];
      if ((kk & pmask) == prefix)
        atomicAdd(&hist[(kk >> shift) & 255u], 1u);
    }
    __syncthreads();

    // Hillis-Steele inclusive scan over the 256 bins.
    const unsigned int v = hist[t];
    scan[t] = v;
    __syncthreads();
#pragma unroll
    for (int off = 1; off < 256; off <<= 1) {
      const unsigned int add = (t >= off) ? scan[t - off] : 0u;
      __syncthreads();
      scan[t] += add;
      __syncthreads();
    }
    const unsigned int incl = scan[t];
    const unsigned int excl = incl - v;

    // Exactly one thread owns the bucket containing rank kRem.
    if (excl <= kRem && kRem < incl) {
      s_prefix = prefix | ((unsigned int)t << shift);
      s_k      = kRem - excl;
    }
    __syncthreads();
    prefix = s_prefix;
    kRem   = s_k;
    __syncthreads();
  }

  if (t == 0) {
    // invert the order-preserving map
    const unsigned int u = prefix;
    const unsigned int orig =
        (u & 0x80000000u) ? (u ^ 0x80000000u) : ~u;
    med[row] = __uint_as_float(orig);
  }
}

// ---------------------------------------------------------------------------
// Kernel 2: out[128,1000] = med[128,768] @ W.T + bias, fp32 WMMA 16x16x4.
// One wave per 16x16 output tile; 8 batch tiles x 63 class tiles = 504 waves
// = 63 blocks x 8 waves. EXEC stays all-ones around every WMMA.
// ---------------------------------------------------------------------------
__global__ __launch_bounds__(256) void gemm_med_kernel(
    const float* __restrict__ med,   // [Bm, C] row-major
    const float* __restrict__ W,     // [NC, C] row-major
    const float* __restrict__ bias,  // [NC]
    float* __restrict__ out,         // [Bm, NC]
    int Bm, int C, int NC) {
  const int lane  = threadIdx.x & 31;
  const int wave  = threadIdx.x >> 5;
  const int nTilesN = (NC + 15) >> 4;                      // 63
  const int tileId  = blockIdx.x * (blockDim.x >> 5) + wave;
  const int bt = tileId / nTilesN;                         // 0..7
  const int nt = tileId % nTilesN;                         // 0..62

  const int half = lane >> 4;        // 0: K={0,1}, 1: K={2,3}
  const int l    = lane & 15;
  const int kOff = half * 2;

  const int rowA = bt * 16 + l;                       // batch row (always < Bm)
  int rowB = nt * 16 + l;                             // class row
  if (rowB >= NC) rowB = NC - 1;                      // clamp: cols >= NC never stored

  const float* aPtr = med + (size_t)rowA * C + kOff;
  const float* bPtr = W   + (size_t)rowB * C + kOff;

  v8f acc = {0.f, 0.f, 0.f, 0.f, 0.f, 0.f, 0.f, 0.f};

#pragma unroll 4
  for (int c = 0; c < C; c += 4) {
    // A fragment: 16x4 f32 — lane l holds A[M=l][K=kOff], A[M=l][K=kOff+1]
    v2f a = *(const v2f*)(aPtr + c);
    // B fragment: 4x16 f32 — lane l holds B[K=kOff][N=l], B[K=kOff+1][N=l]
    v2f b = *(const v2f*)(bPtr + c);
    acc = __builtin_amdgcn_wmma_f32_16x16x4_f32(
        /*neg_a=*/false, a, /*neg_b=*/false, b,
        /*c_mod=*/(short)0, acc, /*reuse_a=*/false, /*reuse_b=*/false);
  }

  // D layout: lane<16 -> N=l, rows M=r; lane>=16 -> N=l, rows M=r+8
  const int nGlob = nt * 16 + l;
  if (nGlob < NC) {
    const float bb = bias[nGlob];
    const int mBase = bt * 16 + half * 8;
#pragma unroll
    for (int r = 0; r < 8; ++r) {
      out[(size_t)(mBase + r) * NC + nGlob] = acc[r] + bb;
    }
  }
}

// ---------------------------------------------------------------------------
extern "C" void kernel_launch(void* const* d_in, const int* in_sizes, int n_in,
                              void* d_out, int out_size, void* d_ws, size_t ws_size,
                              hipStream_t stream) {
  const float* x    = (const float*)d_in[0];   // [128, 768, 28, 28]
  const float* W    = (const float*)d_in[1];   // [1000, 768]
  const float* bias = (const float*)d_in[2];   // [1000]
  float* out = (float*)d_out;                  // [128, 1000]
  float* med = (float*)d_ws;                   // [128*768] scratch

  constexpr int Bm = 128, C = 768, N = 28 * 28, K = N / 2, NC = 1000;

  // Phase 1: 98304 rows, one block each. Reads x once (~308 MB -> ~13 us floor).
  kth_select_kernel<<<Bm * C, 256, 0, stream>>>(x, med, N, N - K);

  // Phase 2: 8 x 63 = 504 wave-tiles = 63 blocks of 8 waves.
  gemm_med_kernel<<<63, 256, 0, stream>>>(med, W, bias, out, Bm, C, NC);
}